// Until_74010876444811
// MI455X (gfx1250) — compile-verified
//
#include <hip/hip_runtime.h>
#include <math.h>

#ifndef __has_builtin
#define __has_builtin(x) 0
#endif

// ---------------------------------------------------------------------------
// Problem constants (reference: B=64, T=8192, X=32, layout [B, T, X], f32)
// ---------------------------------------------------------------------------
#define UB 64
#define UT 8192
#define UX 32
#define LARGE_NUM 1.0e6f

#define CHUNK   128                  // timesteps per chunk (sequential per thread)
#define NCHUNK  (UT / CHUNK)         // 64 chunks
#define SUBS    8                    // chunks handled per 256-thread block
#define TS      16                   // timesteps per slab (per chunk)
#define NSLAB   (CHUNK / TS)         // 8 slabs per chunk
#define ROWF    (TS * UX)            // 512 floats: one chunk's slab segment (contiguous)
#define SLABF   (SUBS * ROWF)        // 4096 floats staged per input per slab (16 KB)

// ---------------------------------------------------------------------------
// CDNA5 Tensor Data Mover path (gfx1250): tensor_load_to_lds + s_wait_tensorcnt
// 2D tile: tile_dim0 = 512 floats (16 t x 32 x contiguous), tile_dim1 = 8 rows
// (one row per chunk), tensor_dim0_stride = 4096 floats (=128 t = chunk pitch).
// ---------------------------------------------------------------------------
#if __has_builtin(__builtin_amdgcn_tensor_load_to_lds) && __has_builtin(__builtin_amdgcn_s_wait_tensorcnt)
#define USE_TDM 1
typedef unsigned int u32x4 __attribute__((ext_vector_type(4)));
typedef int          i32x8 __attribute__((ext_vector_type(8)));
typedef int          i32x4 __attribute__((ext_vector_type(4)));

__device__ __forceinline__ void tdm_load_slab(const float* gsrc, float* ldsDst) {
  unsigned long long ga = (unsigned long long)(uintptr_t)gsrc;
  u32x4 g0;
  g0[0] = 1u;                                           // count=1 (valid user D#)
  g0[1] = (unsigned int)(uintptr_t)ldsDst;              // lds_addr (byte offset, low 32 bits of generic ptr)
  g0[2] = (unsigned int)ga;                             // global_addr[31:0]
  g0[3] = ((unsigned int)(ga >> 32) & 0x01FFFFFFu)      // global_addr[56:32]
          | 0x80000000u;                                // type = 2 ("image")
  i32x8 g1;
  g1[0] = 0x00020000;                                   // data_size = 4B; no multicast/pad/iterate
  g1[1] = 0;                                            // abar=0, tensor_dim0[15:0]=0
  g1[2] = 0x4000;                                       // tensor_dim0 = 1<<30 (no OOB clip); tensor_dim1[15:0]=0
  g1[3] = (ROWF << 16) | 0x4000;                        // tile_dim0 = 512; tensor_dim1 = 1<<30
  g1[4] = SUBS;                                         // tile_dim1 = 8 rows; tile_dim2 = 0
  g1[5] = CHUNK * UX;                                   // tensor_dim0_stride = 4096 floats
  g1[6] = 0;
  g1[7] = 0;
  i32x4 z4 = {0, 0, 0, 0};                              // groups 2/3 unused (2D tile)
#if defined(__clang_major__) && (__clang_major__ >= 23)
  i32x8 z8 = {0, 0, 0, 0, 0, 0, 0, 0};
  __builtin_amdgcn_tensor_load_to_lds(g0, g1, z4, z4, z8, 0);   // clang-23 lane: 6-arg form
#else
  __builtin_amdgcn_tensor_load_to_lds(g0, g1, z4, z4, 0);       // ROCm 7.2 clang-22: 5-arg form
#endif
}
#define TDM_WAIT_MORE() __builtin_amdgcn_s_wait_tensorcnt(2)   // slab s done, s+1 may fly
#define TDM_WAIT_LAST() __builtin_amdgcn_s_wait_tensorcnt(0)
#else
#define USE_TDM 0
#endif

// ---------------------------------------------------------------------------
// Pass 1: per chunk, compose clamp functions -> (L, H).
//   H' = min(hi, max(lo, H)); L' = min(hi, max(lo, L)); hi=phi[t], lo=psi[t].
// Block = 256 threads: sub = tid/32 picks a chunk, x = tid%32 the channel.
// Grid = UB * (NCHUNK / SUBS) = 512 blocks.
// ---------------------------------------------------------------------------
__global__ void __launch_bounds__(256)
until_pass1(const float* __restrict__ phi, const float* __restrict__ psi,
            float* __restrict__ Lw, float* __restrict__ Hw) {
  __shared__ __align__(128) float sPhi[2][SLABF];
  __shared__ __align__(128) float sPsi[2][SLABF];

  const int b   = blockIdx.x >> 3;
  const int g   = blockIdx.x & 7;
  const int tid = threadIdx.x;
  const int sub = tid >> 5;
  const int x   = tid & 31;
  const int chunk = g * SUBS + sub;
  const long long gbase = (long long)b * UT * UX;

#if USE_TDM
  auto stage = [&](int s, int par) {
    if (tid < 32) {  // wave 0 issues DMA for the whole block (TDM ignores EXEC lanes)
      long long eoff = gbase + (long long)(g * SUBS * CHUNK + s * TS) * UX;
      tdm_load_slab(phi + eoff, &sPhi[par][0]);
      tdm_load_slab(psi + eoff, &sPsi[par][0]);
    }
  };
#else
  auto stage = [&](int s, int par) {
#pragma unroll
    for (int it = 0; it < SLABF / (256 * 4); ++it) {   // 4 x float4 per thread per input
      int li  = it * 256 + tid;                        // 0..1023 16B granules
      int ss  = li >> 7;
      int r   = li & 127;
      int tt  = r >> 3;
      int q   = r & 7;
      long long eoff = gbase + (long long)(((g * SUBS + ss) * CHUNK) + s * TS + tt) * UX + q * 4;
      *(float4*)&sPhi[par][li * 4] = *(const float4*)&phi[eoff];
      *(float4*)&sPsi[par][li * 4] = *(const float4*)&psi[eoff];
    }
  };
#endif

  float Lc = -INFINITY;   // identity clamp
  float Hc =  INFINITY;

  stage(0, 0);
  for (int s = 0; s < NSLAB; ++s) {
    const int par = s & 1;
    if (s + 1 < NSLAB) {
      stage(s + 1, par ^ 1);
#if USE_TDM
      if (tid < 32) TDM_WAIT_MORE();
#endif
    } else {
#if USE_TDM
      if (tid < 32) TDM_WAIT_LAST();
#endif
    }
    __syncthreads();
    const float* pphi = &sPhi[par][sub * ROWF];
    const float* ppsi = &sPsi[par][sub * ROWF];
#pragma unroll
    for (int tt = 0; tt < TS; ++tt) {
      float hi = pphi[tt * UX + x];
      float lo = ppsi[tt * UX + x];
      Hc = fminf(hi, fmaxf(lo, Hc));
      Lc = fminf(hi, fmaxf(lo, Lc));
    }
    __syncthreads();
  }

  const int o = (b * NCHUNK + chunk) * UX + x;
  Lw[o] = Lc;
  Hw[o] = Hc;
}

// ---------------------------------------------------------------------------
// Pass 2: sequentially apply chunk clamps to propagate carries (tiny).
// carry_in[k] = state entering chunk k; carry_in[0] = -LARGE.
// ---------------------------------------------------------------------------
__global__ void __launch_bounds__(256)
until_pass2(const float* __restrict__ Lw, const float* __restrict__ Hw,
            float* __restrict__ carry) {
  const int tid = blockIdx.x * 256 + threadIdx.x;     // 0 .. B*X-1
  if (tid >= UB * UX) return;
  const int b = tid >> 5;
  const int x = tid & 31;
  float c = -LARGE_NUM;
  for (int k = 0; k < NCHUNK; ++k) {
    const int o = (b * NCHUNK + k) * UX + x;
    carry[o] = c;
    c = fminf(Hw[o], fmaxf(Lw[o], c));
  }
}

// ---------------------------------------------------------------------------
// Pass 3: re-scan each chunk with its incoming carry, write outputs.
// Same TDM double-buffered staging; inputs mostly hit L2 (192 MB) this pass.
// Output stores are non-temporal: the 64 MiB result is never re-read, so keep
// it from evicting not-yet-read input lines out of L2.
// ---------------------------------------------------------------------------
__global__ void __launch_bounds__(256)
until_pass3(const float* __restrict__ phi, const float* __restrict__ psi,
            const float* __restrict__ carry, float* __restrict__ out) {
  __shared__ __align__(128) float sPhi[2][SLABF];
  __shared__ __align__(128) float sPsi[2][SLABF];

  const int b   = blockIdx.x >> 3;
  const int g   = blockIdx.x & 7;
  const int tid = threadIdx.x;
  const int sub = tid >> 5;
  const int x   = tid & 31;
  const int chunk = g * SUBS + sub;
  const long long gbase = (long long)b * UT * UX;

#if USE_TDM
  auto stage = [&](int s, int par) {
    if (tid < 32) {
      long long eoff = gbase + (long long)(g * SUBS * CHUNK + s * TS) * UX;
      tdm_load_slab(phi + eoff, &sPhi[par][0]);
      tdm_load_slab(psi + eoff, &sPsi[par][0]);
    }
  };
#else
  auto stage = [&](int s, int par) {
#pragma unroll
    for (int it = 0; it < SLABF / (256 * 4); ++it) {
      int li  = it * 256 + tid;
      int ss  = li >> 7;
      int r   = li & 127;
      int tt  = r >> 3;
      int q   = r & 7;
      long long eoff = gbase + (long long)(((g * SUBS + ss) * CHUNK) + s * TS + tt) * UX + q * 4;
      *(float4*)&sPhi[par][li * 4] = *(const float4*)&phi[eoff];
      *(float4*)&sPsi[par][li * 4] = *(const float4*)&psi[eoff];
    }
  };
#endif

  float c = carry[(b * NCHUNK + chunk) * UX + x];

  stage(0, 0);
  for (int s = 0; s < NSLAB; ++s) {
    const int par = s & 1;
    if (s + 1 < NSLAB) {
      stage(s + 1, par ^ 1);
#if USE_TDM
      if (tid < 32) TDM_WAIT_MORE();
#endif
    } else {
#if USE_TDM
      if (tid < 32) TDM_WAIT_LAST();
#endif
    }
    __syncthreads();
    const float* pphi = &sPhi[par][sub * ROWF];
    const float* ppsi = &sPsi[par][sub * ROWF];
    long long obase = gbase + (long long)(chunk * CHUNK + s * TS) * UX + x;
#pragma unroll
    for (int tt = 0; tt < TS; ++tt) {
      float hi = pphi[tt * UX + x];
      float lo = ppsi[tt * UX + x];
      c = fminf(hi, fmaxf(lo, c));                        // out[t] = min(phi, max(psi, out[t-1]))
      __builtin_nontemporal_store(c, &out[obase + (long long)tt * UX]);  // coalesced 128B NT store
    }
    __syncthreads();
  }
}

// ---------------------------------------------------------------------------
// Launch: trace1 = phi, trace2 = psi (setup_inputs order). Workspace layout:
//   Lw[B*NCHUNK*X] | Hw[B*NCHUNK*X] | carry[B*NCHUNK*X]  (3 * 512 KB = 1.5 MB)
// ---------------------------------------------------------------------------
extern "C" void kernel_launch(void* const* d_in, const int* in_sizes, int n_in,
                              void* d_out, int out_size, void* d_ws, size_t ws_size,
                              hipStream_t stream) {
  (void)in_sizes; (void)n_in; (void)out_size; (void)ws_size;
  const float* phi = (const float*)d_in[0];
  const float* psi = (const float*)d_in[1];
  float* out = (float*)d_out;

  const size_t seg = (size_t)UB * NCHUNK * UX;
  float* Lw    = (float*)d_ws;
  float* Hw    = Lw + seg;
  float* carry = Hw + seg;

  const dim3 blk(256);
  const dim3 grid1(UB * (NCHUNK / SUBS));   // 512 blocks
  until_pass1<<<grid1, blk, 0, stream>>>(phi, psi, Lw, Hw);
  until_pass2<<<dim3((UB * UX + 255) / 256), blk, 0, stream>>>(Lw, Hw, carry);
  until_pass3<<<grid1, blk, 0, stream>>>(phi, psi, carry, out);
}